// GATNet_49014166782582
// MI455X (gfx1250) — compile-verified
//
#include <hip/hip_runtime.h>
#include <hip/hip_bf16.h>

typedef float v2f __attribute__((ext_vector_type(2)));
typedef float v8f __attribute__((ext_vector_type(8)));

// ---------------------------------------------------------------------------
// WMMA skinny GEMM: Y[R, N] = Arows[R, K] @ W[K, N]   (fp32, V_WMMA_F32_16X16X4_F32)
// MODE 0: A row r at A + r*K                      (plain row-major)
// MODE 1: A row r=(n,b) at A + (b*nsrc + n)*K     (fused [B,N,C]->[N,B,C] transpose)
// MODE 2: A row r=(d,b) at A + (gidx[d]*4 + b)*K  (fused node gather, B=4)
// One wave32 per 16-row tile; R is always a multiple of 16 here.
// ---------------------------------------------------------------------------
template <int K, int N, int MODE>
__global__ void gemm_wmma(const float* __restrict__ A, const float* __restrict__ W,
                          float* __restrict__ Y, const int* __restrict__ gidx,
                          int R, int nsrc) {
  const int wave = blockIdx.x * (blockDim.x >> 5) + (threadIdx.x >> 5);
  const int lane = threadIdx.x & 31;
  const int row0 = wave * 16;
  if (row0 >= R) return;

  const int m    = lane & 15;        // A: row within tile / B: column
  const int kb   = (lane >> 4) << 1; // 0 or 2: K sub-offset for A/B fragments
  const int r    = row0 + m;

  // Per-lane A row base (fused transpose / gather)
  long off;
  if (MODE == 0) {
    off = (long)r * K;
  } else if (MODE == 1) {
    const int n = r >> 2, b = r & 3;           // B == 4
    off = ((long)b * nsrc + n) * (long)K;
  } else {
    const int d = gidx[r >> 2];
    off = ((long)d * 4 + (r & 3)) * (long)K;
  }
  const float* __restrict__ arow = A + off;

  // Preload B (weight) fragments: [K/4 k-steps][N/16 col-tiles], 2 VGPRs each.
  v2f wf[K / 4][N / 16];
#pragma unroll
  for (int ks = 0; ks < K / 4; ++ks) {
#pragma unroll
    for (int nt = 0; nt < N / 16; ++nt) {
      const int kk = ks * 4 + kb;
      const int nn = nt * 16 + m;
      wf[ks][nt].x = W[kk * N + nn];
      wf[ks][nt].y = W[(kk + 1) * N + nn];
    }
  }

  v8f acc[N / 16] = {};
#pragma unroll
  for (int ks = 0; ks < K / 4; ++ks) {
    v2f a;
    a.x = arow[ks * 4 + kb];
    a.y = arow[ks * 4 + kb + 1];
#pragma unroll
    for (int nt = 0; nt < N / 16; ++nt) {
      acc[nt] = __builtin_amdgcn_wmma_f32_16x16x4_f32(
          /*neg_a=*/false, a, /*neg_b=*/false, wf[ks][nt],
          /*c_mod=*/(short)0, acc[nt], /*reuse_a=*/false, /*reuse_b=*/false);
    }
  }

  // C/D layout: vgpr v holds M = v + 8*(lane>=16), N = lane&15
  const int half = lane >> 4;
  const int cn   = lane & 15;
#pragma unroll
  for (int nt = 0; nt < N / 16; ++nt) {
#pragma unroll
    for (int vr = 0; vr < 8; ++vr) {
      const int rr = row0 + half * 8 + vr;
      Y[(long)rr * N + nt * 16 + cn] = acc[nt][vr];
    }
  }
}

// ---------------------------------------------------------------------------
// Edge kernel: one wave32 per edge.
// msg = ys[src] * sigmoid(yd[dst] * ys[src] * (ew*We)); atomic scatter-add to agg[dst].
// ys/yd/agg rows are B*H contiguous -> fully coalesced; all hot data L2-resident.
// ---------------------------------------------------------------------------
template <int BH, int H>
__global__ void edge_kernel(const float* __restrict__ ys, const float* __restrict__ yd,
                            const int* __restrict__ src, const int* __restrict__ dst,
                            const float* __restrict__ ew, const float* __restrict__ We,
                            float* __restrict__ agg, float* __restrict__ cnt, int E) {
  const int e = blockIdx.x * (blockDim.x >> 5) + (threadIdx.x >> 5);
  if (e >= E) return;
  const int lane = threadIdx.x & 31;
  const int s = src[e], d = dst[e];
  const float w = ew[e];
  const float* __restrict__ ysr = ys + (long)s * BH;
  const float* __restrict__ ydr = yd + (long)d * BH;
  float* __restrict__ ar        = agg + (long)d * BH;
#pragma unroll
  for (int i = lane; i < BH; i += 32) {
    const float xj = ysr[i];
    const float xi = ydr[i];
    const float ev = w * We[i & (H - 1)];
    const float t  = xi * xj * ev;
    const float sg = 1.0f / (1.0f + __expf(-t));
    atomicAdd(ar + i, xj * sg);
  }
  if (lane == 0) atomicAdd(cnt + d, 1.0f);
}

// ---------------------------------------------------------------------------
// Finalize: one wave32 per dst node.
// out = yd + agg/max(cnt,1); per-node BN over (B,C); residual += xd; leaky_relu.
// FINAL=true writes the [B, N2, Cout] transposed final output.
// ---------------------------------------------------------------------------
template <int BH, int H, bool FINAL>
__global__ void finalize_kernel(const float* __restrict__ xs, const int* __restrict__ res,
                                const float* __restrict__ yd, const float* __restrict__ agg,
                                const float* __restrict__ cnt, float* __restrict__ out,
                                int Nd) {
  const int d = blockIdx.x * (blockDim.x >> 5) + (threadIdx.x >> 5);
  if (d >= Nd) return;
  const int lane = threadIdx.x & 31;
  constexpr int PER = BH / 32;

  const float inv_cnt = 1.0f / fmaxf(cnt[d], 1.0f);
  const long  base    = (long)d * BH;
  const long  xbase   = (long)res[d] * BH;  // xd[d] row == xs[res[d]] row (B*H contiguous)

  float vals[PER], xdv[PER];
  float sum = 0.0f, ssq = 0.0f;
#pragma unroll
  for (int j = 0; j < PER; ++j) {
    const int i  = lane + j * 32;
    const float o = yd[base + i] + agg[base + i] * inv_cnt;
    vals[j] = o;
    xdv[j]  = xs[xbase + i];
    sum += o;
    ssq += o * o;
  }
#pragma unroll
  for (int mask = 16; mask >= 1; mask >>= 1) {
    sum += __shfl_xor(sum, mask, 32);
    ssq += __shfl_xor(ssq, mask, 32);
  }
  const float mu  = sum * (1.0f / BH);
  const float var = ssq * (1.0f / BH) - mu * mu;   // biased variance
  const float is  = rsqrtf(var + 1e-5f);

#pragma unroll
  for (int j = 0; j < PER; ++j) {
    const int i = lane + j * 32;
    float v = xdv[j] + (vals[j] - mu) * is;
    v = v > 0.0f ? v : 0.01f * v;   // leaky_relu(0.01)
    if (FINAL) {
      const int b = i / H, c = i & (H - 1);
      out[((long)b * Nd + d) * H + c] = v;   // -> [B, N2, Cout]
    } else {
      out[base + i] = v;                     // -> [N_dst, B, H]
    }
  }
}

// ---------------------------------------------------------------------------
// Launch
// Inputs: 0:X 1:src0 2:dst0 3:ew0 4:res0 5:src1 6:dst1 7:ew1 8:res1
//         9:Wn1 10:We1 11:U1 12:V1 13:Wn2 14:We2 15:U2 16:V2
// ---------------------------------------------------------------------------
extern "C" void kernel_launch(void* const* d_in, const int* in_sizes, int n_in,
                              void* d_out, int out_size, void* d_ws, size_t ws_size,
                              hipStream_t stream) {
  constexpr int B = 4, Cin = 32, H = 16, Cout = 32;
  const float* X    = (const float*)d_in[0];
  const int*   src0 = (const int*)d_in[1];
  const int*   dst0 = (const int*)d_in[2];
  const float* ew0  = (const float*)d_in[3];
  const int*   res0 = (const int*)d_in[4];
  const int*   src1 = (const int*)d_in[5];
  const int*   dst1 = (const int*)d_in[6];
  const float* ew1  = (const float*)d_in[7];
  const int*   res1 = (const int*)d_in[8];
  const float* Wn1  = (const float*)d_in[9];
  const float* We1  = (const float*)d_in[10];
  const float* U1   = (const float*)d_in[11];
  const float* V1   = (const float*)d_in[12];
  const float* Wn2  = (const float*)d_in[13];
  const float* We2  = (const float*)d_in[14];
  const float* U2   = (const float*)d_in[15];
  const float* V2   = (const float*)d_in[16];

  const int E0 = in_sizes[1];
  const int N1 = in_sizes[4];
  const int E1 = in_sizes[5];
  const int N2 = in_sizes[8];
  const int N0 = in_sizes[0] / (B * Cin);

  // Workspace layout (floats)
  float* ws = (float*)d_ws;
  const size_t szXS = (size_t)((N0 * B * H > N1 * B * Cout) ? N0 * B * H : N1 * B * Cout);
  const size_t szYD = (size_t)((N1 * B * H > N2 * B * Cout) ? N1 * B * H : N2 * B * Cout);
  float* xs  = ws;                       // max(N0*B*H, N1*B*Cout)
  float* ys  = xs + szXS;                // same size
  float* yd  = ys + szXS;                // max(N1*B*H, N2*B*Cout)
  float* agg = yd + szYD;                // same size
  float* cnt = agg + szYD;               // max(N1, N2)
  float* h1  = cnt + (size_t)((N1 > N2) ? N1 : N2);  // N1*B*H

  const dim3 blk(256);
  auto waves_grid = [](int tiles) { return dim3((unsigned)((tiles + 7) / 8)); };

  // ------------------------- Layer 1 (Cin=32 -> H=16) -------------------------
  const int R1s = N0 * B;   // source rows
  const int R1d = N1 * B;   // dst rows
  gemm_wmma<32, 16, 1><<<waves_grid(R1s / 16), blk, 0, stream>>>(X,  Wn1, xs, nullptr, R1s, N0);
  gemm_wmma<16, 16, 0><<<waves_grid(R1s / 16), blk, 0, stream>>>(xs, V1,  ys, nullptr, R1s, 0);
  gemm_wmma<16, 16, 2><<<waves_grid(R1d / 16), blk, 0, stream>>>(xs, U1,  yd, res0,    R1d, 0);
  hipMemsetAsync(agg, 0, (size_t)R1d * H * sizeof(float), stream);
  hipMemsetAsync(cnt, 0, (size_t)N1 * sizeof(float), stream);
  edge_kernel<64, 16><<<dim3((unsigned)((E0 + 7) / 8)), blk, 0, stream>>>(
      ys, yd, src0, dst0, ew0, We1, agg, cnt, E0);
  finalize_kernel<64, 16, false><<<dim3((unsigned)((N1 + 7) / 8)), blk, 0, stream>>>(
      xs, res0, yd, agg, cnt, h1, N1);

  // ------------------------- Layer 2 (H=16 -> Cout=32) ------------------------
  const int R2s = N1 * B;
  const int R2d = N2 * B;
  gemm_wmma<16, 32, 0><<<waves_grid(R2s / 16), blk, 0, stream>>>(h1, Wn2, xs, nullptr, R2s, 0);
  gemm_wmma<32, 32, 0><<<waves_grid(R2s / 16), blk, 0, stream>>>(xs, V2,  ys, nullptr, R2s, 0);
  gemm_wmma<32, 32, 2><<<waves_grid(R2d / 16), blk, 0, stream>>>(xs, U2,  yd, res1,    R2d, 0);
  hipMemsetAsync(agg, 0, (size_t)R2d * Cout * sizeof(float), stream);
  hipMemsetAsync(cnt, 0, (size_t)N2 * sizeof(float), stream);
  edge_kernel<128, 32><<<dim3((unsigned)((E1 + 7) / 8)), blk, 0, stream>>>(
      ys, yd, src1, dst1, ew1, We2, agg, cnt, E1);
  finalize_kernel<128, 32, true><<<dim3((unsigned)((N2 + 7) / 8)), blk, 0, stream>>>(
      xs, res1, yd, agg, cnt, (float*)d_out, N2);
}